// Downsample_26766236189271
// MI455X (gfx1250) — compile-verified
//
#include <hip/hip_runtime.h>
#include <float.h>

typedef unsigned int u32;
typedef unsigned long long u64;

#define NB   8        // batches
#define NPTS 65536    // points per batch
#define NS   4096     // samples
#define NC   64       // feature channels
#define WPB  16       // workgroups per batch
#define PTS_WG (NPTS / WPB)   // 4096 points owned per WG
#define NT   256              // threads per WG (8 waves, wave32)
#define PPT  (PTS_WG / NT)    // 16 points per thread
#define NWAVES (NT / 32)

#define SLOT_BYTES 32
#define SLOTS_BYTES (NB * WPB * 2 * SLOT_BYTES)   // parity double-buffered: 8 KB

// 128-bit chunk type for the async global->LDS DMA builtin
typedef int v4i __attribute__((ext_vector_type(4)));
typedef __attribute__((address_space(1))) v4i* gv4p;   // global
typedef __attribute__((address_space(3))) v4i* lv4p;   // LDS

// ---- DEV(agent)-scope L2 accessors: loads bypass WGP$ (scope > cache scope) ----
__device__ __forceinline__ u64 ld_rlx_u64(u64* p) {
  return __hip_atomic_load(p, __ATOMIC_RELAXED, __HIP_MEMORY_SCOPE_AGENT);
}
__device__ __forceinline__ void st_rel_u64(u64* p, u64 v) {
  __hip_atomic_store(p, v, __ATOMIC_RELEASE, __HIP_MEMORY_SCOPE_AGENT);
}
__device__ __forceinline__ float ld_rlx_f32(float* p) {
  return __hip_atomic_load(p, __ATOMIC_RELAXED, __HIP_MEMORY_SCOPE_AGENT);
}
__device__ __forceinline__ void st_rlx_f32(float* p, float v) {
  __hip_atomic_store(p, v, __ATOMIC_RELAXED, __HIP_MEMORY_SCOPE_AGENT);
}

// key-only argmax reduce over wave32; key = [dist_bits:32][~idx] (dist >= 0 so
// float bit-pattern order == numeric order; inverted idx = first-index ties)
__device__ __forceinline__ u64 wave_red_key(u64 k) {
#pragma unroll
  for (int m = 1; m < 32; m <<= 1) {
    u64 o = __shfl_xor(k, m, 32);
    k = (o > k) ? o : k;
  }
  return k;
}

__global__ __launch_bounds__(NT)
void fps_kernel(const float* __restrict__ p, char* __restrict__ ws) {
  const int b    = blockIdx.x / WPB;     // batch
  const int w    = blockIdx.x % WPB;     // workgroup-within-batch
  const int tid  = threadIdx.x;
  const int lane = tid & 31;
  const int wv   = tid >> 5;
  const int base = w * PTS_WG;

  __shared__ float s_stage[PTS_WG * 3];          // 48 KB coord slice (stays live)
  __shared__ u64   s_wkey[NWAVES];
  __shared__ float s_q[3];

  int* idx_out = (int*)(ws + SLOTS_BYTES);       // int[NB][NS]

  // ---- stage this WG's contiguous 48 KB coord slice into LDS (async DMA) ----
  const float* gsrc = p + ((size_t)b * NPTS + base) * 3;   // 16B-aligned slice
#if __has_builtin(__builtin_amdgcn_global_load_async_to_lds_b128)
  {
    gv4p gp = (gv4p)gsrc;
    lv4p lp = (lv4p)&s_stage[0];
#pragma unroll
    for (int c = 0; c < (PTS_WG * 12) / (16 * NT); ++c) {   // 12 chunks of 16B/thread
      int ci = c * NT + tid;
      __builtin_amdgcn_global_load_async_to_lds_b128(gp + ci, lp + ci, 0, 0);
    }
  }
#if __has_builtin(__builtin_amdgcn_s_wait_asynccnt)
  __builtin_amdgcn_s_wait_asynccnt(0);
#else
  asm volatile("s_wait_asynccnt 0" ::: "memory");
#endif
#else
  for (int c = tid; c < PTS_WG * 3; c += NT) s_stage[c] = gsrc[c];
#endif
  __syncthreads();

  // park coords + running min-distance in registers (16 pts/thread, strided by NT)
  float px[PPT], py[PPT], pz[PPT], dist[PPT];
#pragma unroll
  for (int j = 0; j < PPT; ++j) {
    int l = j * NT + tid;
    px[j] = s_stage[l * 3 + 0];
    py[j] = s_stage[l * 3 + 1];
    pz[j] = s_stage[l * 3 + 2];
    dist[j] = FLT_MAX;
  }

  // first query = point 0 of the batch (PointNet++ convention)
  float qx = p[(size_t)b * NPTS * 3 + 0];
  float qy = p[(size_t)b * NPTS * 3 + 1];
  float qz = p[(size_t)b * NPTS * 3 + 2];
  if (w == 0 && tid == 0) idx_out[b * NS + 0] = 0;

  for (int it = 1; it < NS; ++it) {
    // ---- local distance update + running argmax (value + index only) ----
    float bestv = -1.0f;
    u32 bg = 0;
#pragma unroll
    for (int j = 0; j < PPT; ++j) {
      float dx = px[j] - qx, dy = py[j] - qy, dz = pz[j] - qz;
      float d  = __fmaf_rn(dx, dx, __fmaf_rn(dy, dy, dz * dz));
      float nd = fminf(dist[j], d);
      dist[j] = nd;
      if (nd > bestv) { bestv = nd; bg = (u32)(base + j * NT + tid); }
    }
    u64 key = ((u64)__float_as_uint(bestv) << 32) | (u64)(0xFFFFFFFFu - bg);
    key = wave_red_key(key);
    if (lane == 0) s_wkey[wv] = key;
    __syncthreads();

    if (wv == 0) {
      // ---- cross-wave reduce to WG best (key only) ----
      u64 k2 = (lane < NWAVES) ? s_wkey[lane] : 0ull;
      k2 = wave_red_key(k2);

      // ---- publish (coords from LDS, tagged key) to this WG's parity slot ----
      char* myslot = ws + (size_t)(((b * WPB + w) * 2 + (it & 1)) * SLOT_BYTES);
      if (lane == 0) {
        u32 gi  = 0xFFFFFFFFu - (u32)(k2 & 0xFFFFFFFFu);         // WG-best index
        int loc = (int)gi - base;                                // within our slice
        u64 Bv  = (k2 & 0xFFFFFFFF00000000ull)
                | ((u64)((0xFFFFu - gi) & 0xFFFFu) << 16)        // ~idx (16b)
                | (u64)(it & 0xFFF);                              // freshness tag
        st_rlx_f32((float*)(myslot + 0), s_stage[loc * 3 + 0]);
        st_rlx_f32((float*)(myslot + 4), s_stage[loc * 3 + 1]);
        st_rlx_f32((float*)(myslot + 8), s_stage[loc * 3 + 2]);
        st_rel_u64((u64*)(myslot + 16), Bv);   // release: coords visible first
      }

      // ---- poll all 16 slots of this batch (lane i -> slot i) ----
      bool act = (lane < WPB);
      int  sw  = act ? lane : 0;
      char* sl = ws + (size_t)(((b * WPB + sw) * 2 + (it & 1)) * SLOT_BYTES);
      u64 Bl = 0;
      for (;;) {
        if (act) Bl = ld_rlx_u64((u64*)(sl + 16));
        bool fresh = !act || ((u32)(Bl & 0xFFFu) == (u32)(it & 0xFFF));
        if (__all(fresh)) break;
        __builtin_amdgcn_s_sleep(0);
      }
      u64 kk = wave_red_key(act ? Bl : 0ull);   // global winner key (all lanes)
      if (lane == 0) {
        u32 gi = 0xFFFFu - ((u32)(kk >> 16) & 0xFFFFu);
        int ww = (int)(gi >> 12);               // winning WG = gi / PTS_WG
        char* slw = ws + (size_t)(((b * WPB + ww) * 2 + (it & 1)) * SLOT_BYTES);
        s_q[0] = ld_rlx_f32((float*)(slw + 0));
        s_q[1] = ld_rlx_f32((float*)(slw + 4));
        s_q[2] = ld_rlx_f32((float*)(slw + 8));
        if (w == 0) idx_out[b * NS + it] = (int)gi;
      }
    }
    __syncthreads();
    qx = s_q[0]; qy = s_q[1]; qz = s_q[2];
  }
}

// ---- gather p_s: [B,S,3] ----
__global__ __launch_bounds__(256)
void gather_p_kernel(const float* __restrict__ p, const int* __restrict__ idx,
                     float* __restrict__ out) {
  int i = blockIdx.x * blockDim.x + threadIdx.x;   // 0 .. NB*NS-1
  if (i >= NB * NS) return;
  int b = i >> 12;
  int gi = idx[i];
  const float* src = p + ((size_t)b * NPTS + (size_t)gi) * 3;
  float* dst = out + (size_t)i * 3;
  dst[0] = src[0]; dst[1] = src[1]; dst[2] = src[2];
}

// ---- gather x_s: [B,C,S]; 4 samples/thread -> b128 coalesced writes ----
typedef float v4f __attribute__((ext_vector_type(4)));
__global__ __launch_bounds__(256)
void gather_x_kernel(const float* __restrict__ x, const int* __restrict__ idx,
                     float* __restrict__ out) {
  size_t t = (size_t)blockIdx.x * blockDim.x + threadIdx.x;  // 0 .. NB*NC*NS/4-1
  if (t >= (size_t)NB * NC * NS / 4) return;
  int s4 = (int)(t & (NS / 4 - 1)) * 4;
  int c  = (int)((t >> 10) & (NC - 1));
  int b  = (int)(t >> 16);
  const v4i gi = *(const v4i*)&idx[b * NS + s4];
  const float* row = x + ((size_t)b * NC + c) * NPTS;
  v4f v;
  v.x = row[gi.x]; v.y = row[gi.y]; v.z = row[gi.z]; v.w = row[gi.w];
  *(v4f*)&out[(((size_t)b * NC + c) * NS) + s4] = v;
}

extern "C" void kernel_launch(void* const* d_in, const int* in_sizes, int n_in,
                              void* d_out, int out_size, void* d_ws, size_t ws_size,
                              hipStream_t stream) {
  (void)in_sizes; (void)n_in; (void)out_size; (void)ws_size;
  const float* p = (const float*)d_in[0];   // [8, 65536, 3]
  const float* x = (const float*)d_in[1];   // [8, 64, 65536]
  char* ws = (char*)d_ws;
  float* out = (float*)d_out;               // p_s (98304) ++ x_s (2097152)
  const int* idx = (const int*)(ws + SLOTS_BYTES);

  fps_kernel<<<NB * WPB, NT, 0, stream>>>(p, ws);
  gather_p_kernel<<<(NB * NS + 255) / 256, 256, 0, stream>>>(p, idx, out);
  gather_x_kernel<<<((size_t)NB * NC * NS / 4 + 255) / 256, 256, 0, stream>>>(
      x, idx, out + (size_t)NB * NS * 3);
}